// PiecewiseHawkesIntensity_34153579937861
// MI455X (gfx1250) — compile-verified
//
#include <hip/hip_runtime.h>
#include <cstdint>

// Problem constants (from reference): B=16, M=32, P=16, L=1024, L_EVAL=2048
#define B_ 16
#define M_ 32
#define P_ 16
#define L_ 1024
#define LE_ 2048
#define SPLIT 4                 // query chunks per (b,p) path
#define QPB (LE_ / SPLIT)       // 512 queries per block
#define NT 256                  // 8 wave32 waves per block

__global__ __launch_bounds__(NT) void hawkes_intensity_kernel(
    const float* __restrict__ qt,    // [B,P,LE]
    const float* __restrict__ ev,    // [B,P,L]  sorted along last dim
    const float* __restrict__ mu,    // [B,M,P,L]
    const float* __restrict__ al,    // [B,M,P,L]
    const float* __restrict__ be,    // [B,M,P,L]
    const float* __restrict__ nc,    // [B]
    float* __restrict__ out)         // [B,M,P,LE]
{
  __shared__ float s_ev[L_];     // 4 KB: event times for this (b,p) path
  __shared__ float s_dt[QPB];    // 2 KB: delta_t per query (m-invariant)
  __shared__ int   s_idx[QPB];   // 2 KB: gather index per query (m-invariant)

  const int tid    = threadIdx.x;
  const int blk    = blockIdx.x;
  const int qchunk = blk % SPLIT;
  const int bp     = blk / SPLIT;      // b*P + p
  const int p      = bp % P_;
  const int b      = bp / P_;

  const float* evp = ev + (size_t)bp * L_;

  // ---- Stage event times into LDS with CDNA5 async global->LDS DMA ----
  // 1024 floats = 256 x b128 transfers, one per lane/thread.
  // GLOBAL_LOAD_ASYNC_TO_LDS_B128: VDST = LDS byte address (VGPR),
  // VADDR = 64-bit global address (VGPR pair), no SADDR ("off"), GV mode.
  // LDS byte address = low 32 bits of the generic __shared__ pointer
  // (aperture mapping, ISA 10.2). Tracked by ASYNCcnt.
#if defined(__gfx1250__)
  {
    uint32_t lds_addr = (uint32_t)(uintptr_t)(&s_ev[tid * 4]);
    uint64_t gaddr    = (uint64_t)(uintptr_t)(evp + (size_t)tid * 4);
    asm volatile("global_load_async_to_lds_b128 %0, %1, off\n\t"
                 "s_wait_asynccnt 0"
                 :
                 : "v"(lds_addr), "v"(gaddr)
                 : "memory");
  }
#else
  {
    float4 v = ((const float4*)evp)[tid];
    ((float4*)s_ev)[tid] = v;
  }
#endif
  __syncthreads();

  const float ncb    = nc[b];
  const float inv_nc = 1.0f / ncb;
  const int   q0     = qchunk * QPB;

  // ---- Phase 1: branchless binary search (searchsorted 'left') per query ----
  const float* qtp = qt + (size_t)bp * LE_ + q0;
  for (int q = tid; q < QPB; q += NT) {
    const float qn = qtp[q] / ncb;      // exact division to match reference
    int lo = 0;                          // lower_bound over exactly 1024 elems
    #pragma unroll
    for (int s = L_ >> 1; s > 0; s >>= 1)
      lo += (s_ev[lo + s - 1] < qn) ? s : 0;
    const int last = lo - 1;
    const int idx  = (last < 0) ? 0 : last;
    const float tl = (last < 0) ? 0.0f : s_ev[idx];
    s_dt[q]  = qn - tl;
    s_idx[q] = idx;
  }
  __syncthreads();

  // ---- Phase 2: heads outer so each head's 12KB param slab stays cached;
  //      stores are fully coalesced (consecutive q across lanes).
  //      Unroll heads x2 so 12 gathers + 4 stores are in flight. ----
  #pragma unroll 2
  for (int m = 0; m < M_; ++m) {
    const size_t base = ((size_t)(b * M_ + m) * P_ + p) * (size_t)L_;
    const float* __restrict__ mup = mu + base;
    const float* __restrict__ alp = al + base;
    const float* __restrict__ bep = be + base;
    float* __restrict__ outp =
        out + ((size_t)(b * M_ + m) * P_ + p) * (size_t)LE_ + q0;
    #pragma unroll
    for (int q = tid; q < QPB; q += NT) {
      const int   idx = s_idx[q];
      const float dt  = s_dt[q];
      const float muv = mup[idx];
      const float alv = alp[idx];
      const float bev = bep[idx];
      const float e   = __expf(-bev * dt);   // v_exp_f32 path; ~3 ulp
      outp[q] = (muv + (alv - muv) * e) * inv_nc;
    }
  }
}

extern "C" void kernel_launch(void* const* d_in, const int* in_sizes, int n_in,
                              void* d_out, int out_size, void* d_ws, size_t ws_size,
                              hipStream_t stream) {
  (void)in_sizes; (void)n_in; (void)d_ws; (void)ws_size; (void)out_size;
  const float* qt = (const float*)d_in[0];  // query_times  [B,P,LE]
  const float* ev = (const float*)d_in[1];  // event_times  [B,P,L]
  const float* mu = (const float*)d_in[2];  // mu           [B,M,P,L]
  const float* al = (const float*)d_in[3];  // alpha        [B,M,P,L]
  const float* be = (const float*)d_in[4];  // beta         [B,M,P,L]
  const float* nc = (const float*)d_in[5];  // norm_constants [B]
  float* out = (float*)d_out;               // [B,M,P,LE]

  dim3 grid(B_ * P_ * SPLIT);               // 1024 blocks
  dim3 block(NT);                           // 256 threads = 8 wave32
  hawkes_intensity_kernel<<<grid, block, 0, stream>>>(qt, ev, mu, al, be, nc, out);
}